// ByteMemoryBank_8186207666947
// MI455X (gfx1250) — compile-verified
//
#include <hip/hip_runtime.h>
#include <cstdint>
#include <cstddef>

// ---------------------------------------------------------------------------
// ByteMemoryBank for MI455X (gfx1250, wave32)
//
// Memory-bound scatter + EMA + gather. ~1.57 GB min traffic -> ~67us @ 23.3TB/s.
// No matmul structure -> WMMA inapplicable; CDNA5 path used: async global<->LDS
// DMA (ASYNCcnt) for the bank-row gather, native f32 global atomics for scatter.
// ---------------------------------------------------------------------------

#define N_SLOTS_C (1u << 20)   // 1048576
#define D_STATE_C 64u
#define N_POS_C   (1u << 21)   // 2097152
#define MOMENTUM_C 0.9f

#define AS1 __attribute__((address_space(1)))
#define AS3 __attribute__((address_space(3)))

typedef int v2i __attribute__((ext_vector_type(2)));   // b64 async payload type

#if __has_builtin(__builtin_amdgcn_global_load_async_to_lds_b64) && \
    __has_builtin(__builtin_amdgcn_global_store_async_from_lds_b64) && \
    __has_builtin(__builtin_amdgcn_s_wait_asynccnt)
#define USE_ASYNC_BUILTINS 1
#else
#define USE_ASYNC_BUILTINS 0
#endif

// addr = (b0*2^24 + b1*2^16 + b2*2^8 + b3) mod 2^20  -> b0 vanishes entirely.
__device__ __forceinline__ unsigned slot_addr(int4 b) {
    return (((unsigned)b.y & 0xFu) << 16) |
           (((unsigned)b.z & 0xFFu) << 8) |
           ((unsigned)b.w & 0xFFu);
}

// ---------------------------------------------------------------------------
// Kernel 1: scatter. One wave32 per position; lane handles 2 floats of the row.
// ---------------------------------------------------------------------------
__global__ __launch_bounds__(256) void bmb_scatter(
    const int4* __restrict__ bw,        // [N_POS] packed 4x int32
    const float2* __restrict__ states2, // [N_POS * 32]
    float* __restrict__ sums,           // [N_SLOTS * 64] accumulator (pre-zeroed)
    int* __restrict__ hits)             // [N_SLOTS] accumulator (pre-zeroed)
{
    const unsigned wave = threadIdx.x >> 5;
    const unsigned lane = threadIdx.x & 31u;
    const unsigned pos  = blockIdx.x * 8u + wave;

    const int4 b = bw[pos];                       // uniform per wave, coalesced
    const unsigned addr = slot_addr(b);

    const float2 s = states2[(size_t)pos * 32u + lane];

    float* sp = sums + (size_t)addr * D_STATE_C + lane * 2u;
    // Relaxed agent-scope -> non-returning global_atomic_add_f32 (STOREcnt path)
    (void)__hip_atomic_fetch_add(sp + 0, s.x, __ATOMIC_RELAXED, __HIP_MEMORY_SCOPE_AGENT);
    (void)__hip_atomic_fetch_add(sp + 1, s.y, __ATOMIC_RELAXED, __HIP_MEMORY_SCOPE_AGENT);

    if (lane == 0) {
        (void)__hip_atomic_fetch_add(hits + addr, 1, __ATOMIC_RELAXED, __HIP_MEMORY_SCOPE_AGENT);
    }
}

// ---------------------------------------------------------------------------
// Kernel 2: blend. One thread per float4 of each slot (16 threads/slot).
// Reads sums in place, writes new_bank in place (single owner per float4).
// ---------------------------------------------------------------------------
__global__ __launch_bounds__(256) void bmb_blend(
    float4* __restrict__ bank_io,        // [N_SLOTS*16] float4: in=sums, out=new_bank
    const float4* __restrict__ bank_old, // [N_SLOTS*16] float4 (input bank)
    const int* __restrict__ counts_in,   // [N_SLOTS]
    const int* __restrict__ hits)        // [N_SLOTS] (raw hit counts)
{
    const size_t tid  = (size_t)blockIdx.x * 256u + threadIdx.x;
    const unsigned slot = (unsigned)(tid >> 4);
    const size_t idx  = tid;             // slot*16 + quad

    const int h = hits[slot];
    const float4 o = bank_old[idx];
    float4 r = o;
    if (h > 0) {
        const float4 s = bank_io[idx];
        const float alpha = (counts_in[slot] == 0) ? 0.0f : MOMENTUM_C;
        const float beta  = (1.0f - alpha) / (float)h; // mean fold-in
        r.x = alpha * o.x + beta * s.x;
        r.y = alpha * o.y + beta * s.y;
        r.z = alpha * o.z + beta * s.z;
        r.w = alpha * o.w + beta * s.w;
    }
    bank_io[idx] = r;
}

// ---------------------------------------------------------------------------
// Kernel 3: counts. new_counts = counts_in + hits, in place over hits region.
// Runs strictly after blend (stream order) so no reader of raw hits remains.
// ---------------------------------------------------------------------------
__global__ __launch_bounds__(256) void bmb_counts(
    int* __restrict__ counts_io, const int* __restrict__ counts_in)
{
    const unsigned i = blockIdx.x * 256u + threadIdx.x;
    counts_io[i] = counts_in[i] + counts_io[i];
}

// ---------------------------------------------------------------------------
// Kernel 4: gather out = new_bank[addr]. One wave32 per position.
// 256B row copied via async DMA: global -> LDS -> global, no VGPR data path.
// Each lane moves 8 bytes (b64), 32 lanes cover the 256B row.
// ---------------------------------------------------------------------------
__global__ __launch_bounds__(256) void bmb_gather(
    const int4* __restrict__ bw,
    const float* __restrict__ bank_new,
    float* __restrict__ out_states)
{
    __shared__ float lds_buf[8 * D_STATE_C];   // 2 KB: 8 waves x 256B row

    const unsigned wave = threadIdx.x >> 5;
    const unsigned lane = threadIdx.x & 31u;
    const unsigned pos  = blockIdx.x * 8u + wave;

    const int4 b = bw[pos];
    const unsigned addr = slot_addr(b);

    const float* src = bank_new + (size_t)addr * D_STATE_C + lane * 2u;
    float*       dst = out_states + (size_t)pos * D_STATE_C + lane * 2u;
    float*       lp  = &lds_buf[wave * D_STATE_C + lane * 2u];

#if USE_ASYNC_BUILTINS
    __builtin_amdgcn_global_load_async_to_lds_b64(
        (AS1 v2i*)(void*)src, (AS3 v2i*)(void*)lp, 0, 0);
    __builtin_amdgcn_s_wait_asynccnt(0);
    __builtin_amdgcn_global_store_async_from_lds_b64(
        (AS1 v2i*)(void*)dst, (AS3 v2i*)(void*)lp, 0, 0);
    // s_endpgm performs an implicit wait-idle for the outstanding async store.
#else
    const unsigned ldsoff = (unsigned)(uintptr_t)(AS3 char*)lp;
    asm volatile("global_load_async_to_lds_b64 %0, %1, off"
                 :: "v"(ldsoff), "v"((uint64_t)(uintptr_t)src) : "memory");
    asm volatile("s_wait_asynccnt 0" ::: "memory");
    asm volatile("global_store_async_from_lds_b64 %0, %1, off"
                 :: "v"((uint64_t)(uintptr_t)dst), "v"(ldsoff) : "memory");
#endif
}

// ---------------------------------------------------------------------------
// Launch
// ---------------------------------------------------------------------------
extern "C" void kernel_launch(void* const* d_in, const int* in_sizes, int n_in,
                              void* d_out, int out_size, void* d_ws, size_t ws_size,
                              hipStream_t stream) {
    (void)in_sizes; (void)n_in; (void)out_size; (void)d_ws; (void)ws_size;

    const int4*   bw      = (const int4*)d_in[0];   // byte_window [N_POS,4] i32
    const float*  states  = (const float*)d_in[1];  // [N_POS,64] f32
    const float*  bank    = (const float*)d_in[2];  // [N_SLOTS,64] f32
    const int*    counts  = (const int*)d_in[3];    // [N_SLOTS] i32

    // d_out layout (tuple flattened in return order):
    //   out       [N_POS  * 64] f32
    //   new_bank  [N_SLOTS* 64] f32   (doubles as the scatter sum accumulator)
    //   new_counts[N_SLOTS]     i32   (doubles as the hit-count accumulator)
    float* out_states = (float*)d_out;
    float* bank_out   = out_states + (size_t)N_POS_C * D_STATE_C;
    int*   counts_out = (int*)(bank_out + (size_t)N_SLOTS_C * D_STATE_C);

    // Zero accumulators every call (harness does not re-poison between replays).
    (void)hipMemsetAsync(bank_out,   0, (size_t)N_SLOTS_C * D_STATE_C * sizeof(float), stream);
    (void)hipMemsetAsync(counts_out, 0, (size_t)N_SLOTS_C * sizeof(int), stream);

    // 1) scatter: N_POS waves, 8 waves (256 threads) per block
    bmb_scatter<<<N_POS_C / 8u, 256, 0, stream>>>(
        bw, (const float2*)states, bank_out, counts_out);

    // 2) blend: N_SLOTS*16 threads (float4 granularity)
    bmb_blend<<<(N_SLOTS_C * 16u) / 256u, 256, 0, stream>>>(
        (float4*)bank_out, (const float4*)bank, counts, counts_out);

    // 3) counts: one thread per slot, in-place after blend
    bmb_counts<<<N_SLOTS_C / 256u, 256, 0, stream>>>(counts_out, counts);

    // 4) gather via async global<->LDS DMA
    bmb_gather<<<N_POS_C / 8u, 256, 0, stream>>>(bw, bank_out, out_states);
}